// CrossNet_26285199851593
// MI455X (gfx1250) — compile-verified
//
#include <hip/hip_runtime.h>
#include <stdint.h>

// out[b,d] = initial[b,d] * dot(X[b,:], alphas) + X[b,d] + bias[d]
// Bandwidth-bound (0.33 flop/byte): one-pass tiling, X read from HBM exactly once.
// CDNA5 paths: global_load_async_to_lds_b128 (ASYNCcnt DMA) + V_WMMA_F32_16X16X4_F32.

typedef __attribute__((ext_vector_type(2))) float v2f;
typedef __attribute__((ext_vector_type(4))) float v4f;
typedef __attribute__((ext_vector_type(8))) float v8f;

#define DIM 2048
#define TILE_ROWS 16
#define THREADS 256            // 8 waves (wave32)
#define PAD_STRIDE 2052        // 2048 + 4 pad floats: stride%64==4 -> WMMA-A ds_load_b64 conflict-free
#define VEC4_PER_ROW (DIM / 4)                   // 512
#define VEC4_PER_TILE (TILE_ROWS * DIM / 4)      // 8192
#define ITERS (VEC4_PER_TILE / THREADS)          // 32

__global__ __launch_bounds__(THREADS) void crossnet_kernel(
    const float* __restrict__ initial, const float* __restrict__ X,
    const float* __restrict__ alphas, const float* __restrict__ bias,
    float* __restrict__ out)
{
    __shared__ float xsLds[TILE_ROWS * PAD_STRIDE];   // 131,328 B
    __shared__ float alphasLds[DIM];                  //   8,192 B
    __shared__ float biasLds[DIM];                    //   8,192 B
    __shared__ float partialLds[8 * TILE_ROWS];       //     512 B
    __shared__ float scaleLds[TILE_ROWS];             //      64 B

    const int t = (int)threadIdx.x;
    const size_t tileBase = (size_t)blockIdx.x * (size_t)(TILE_ROWS * DIM);
    const float* Xtile = X + tileBase;

    // ---------- Phase 1: async DMA X tile (contiguous 128 KB) into LDS ----------
    #pragma unroll
    for (int k = 0; k < ITERS; ++k) {
        int v   = k * THREADS + t;              // float4 index within tile
        int row = v >> 9;                       // / VEC4_PER_ROW
        int c4  = v & (VEC4_PER_ROW - 1);
        unsigned ldsAddr = (unsigned)(uintptr_t)&xsLds[row * PAD_STRIDE + c4 * 4];
        unsigned gOff    = (unsigned)v * 16u;   // byte offset from tile base
        asm volatile("global_load_async_to_lds_b128 %0, %1, %2"
                     :: "v"(ldsAddr), "v"(gOff), "s"(Xtile) : "memory");
    }
    for (int i = t; i < DIM; i += THREADS) {    // stage alphas + bias (8 KB each)
        alphasLds[i] = alphas[i];
        biasLds[i]   = bias[i];
    }
    asm volatile("s_wait_asynccnt 0" ::: "memory");
    __syncthreads();

    // ---------- Phase 2: scale[r] = dot(X[r,:], alphas) via v_wmma_f32_16x16x4_f32 ----------
    // A (16x4 f32): lane L holds row L%16, K pair {2*(L>>4), 2*(L>>4)+1} of the k0 chunk.
    // B (4x16 f32): same per-lane K pair, value alphas[k] broadcast to all 16 columns,
    //               so every column of D carries the same dot product.
    const int lane = t & 31;
    const int wave = t >> 5;
    const int row  = lane & 15;
    const int hi   = lane >> 4;
    const float* aRow = &xsLds[row * PAD_STRIDE];
    const int kBase = wave * (DIM / 8);          // 256 K-values per wave
    v8f acc = {0.f, 0.f, 0.f, 0.f, 0.f, 0.f, 0.f, 0.f};
    #pragma unroll 4
    for (int i = 0; i < (DIM / 8) / 4; ++i) {    // 64 chained WMMAs per wave
        int k0 = kBase + i * 4;
        v2f a = *(const v2f*)&aRow[k0 + 2 * hi];
        v2f b = *(const v2f*)&alphasLds[k0 + 2 * hi];
        acc = __builtin_amdgcn_wmma_f32_16x16x4_f32(
                  false, a, false, b, (short)0, acc, false, false);
    }
    // D layout: lane L, element v -> row (v + 8*(L>>4)); take column N==0 (lanes 0 and 16).
    if (row == 0) {
        #pragma unroll
        for (int v = 0; v < 8; ++v)
            partialLds[wave * TILE_ROWS + hi * 8 + v] = acc[v];
    }
    __syncthreads();
    if (t < TILE_ROWS) {                         // reduce 8 wave-partials per row
        float s = 0.f;
        #pragma unroll
        for (int w = 0; w < 8; ++w) s += partialLds[w * TILE_ROWS + t];
        scaleLds[t] = s;
    }
    __syncthreads();

    // ---------- Phase 3: out = initial*scale + X + bias (pure streaming) ----------
    const float* iniTile = initial + tileBase;
    float* outTile = out + tileBase;
    #pragma unroll 4
    for (int k = 0; k < ITERS; ++k) {
        int v   = k * THREADS + t;
        int r   = v >> 9;
        int c4  = v & (VEC4_PER_ROW - 1);
        v4f xi  = *(const v4f*)&xsLds[r * PAD_STRIDE + c4 * 4];
        v4f bi  = *(const v4f*)&biasLds[c4 * 4];
        v4f ini = *(const v4f*)&iniTile[(size_t)v * 4];
        float s = scaleLds[r];
        v4f o;
        o.x = ini.x * s + xi.x + bi.x;
        o.y = ini.y * s + xi.y + bi.y;
        o.z = ini.z * s + xi.z + bi.z;
        o.w = ini.w * s + xi.w + bi.w;
        *(v4f*)&outTile[(size_t)v * 4] = o;
    }
}

extern "C" void kernel_launch(void* const* d_in, const int* in_sizes, int n_in,
                              void* d_out, int out_size, void* d_ws, size_t ws_size,
                              hipStream_t stream) {
    (void)n_in; (void)d_ws; (void)ws_size; (void)out_size;
    const float* initial = (const float*)d_in[0];
    const float* X       = (const float*)d_in[1];
    const float* alphas  = (const float*)d_in[2];
    const float* bias    = (const float*)d_in[3];
    float* out = (float*)d_out;

    const int rows = in_sizes[0] / DIM;          // 16384
    dim3 grid(rows / TILE_ROWS);                 // 1024 workgroups
    crossnet_kernel<<<grid, THREADS, 0, stream>>>(initial, X, alphas, bias, out);
}